// TriplaneDensityField_38706245271916
// MI455X (gfx1250) — compile-verified
//
#include <hip/hip_runtime.h>
#include <math.h>

// ---------------------------------------------------------------------------
// Triplane density field for MI455X (gfx1250, wave32).
//
//   out[n] = softplus( mean_r( s0[n,r] * s1[n,r] * s2[n,r] ) )
//
// Strategy:
//   Pass 1: transpose each grid [32,256,256] (CHW) -> [256,256,32] (HWC) so a
//           corner's 32-rank vector is one contiguous 128B cache line.
//   Pass 2: wave32 handles 16 points. Lane m (m<16) owns point m ranks
//           {4k,4k+1}; lane m+16 owns the same point's ranks {4k+2,4k+3} --
//           exactly the documented f32 16x4 WMMA A-matrix layout. The rank
//           reduction is 8 chained V_WMMA_F32_16X16X4_F32 with B = ones.
// ---------------------------------------------------------------------------

typedef __attribute__((ext_vector_type(2))) float v2f;
typedef __attribute__((ext_vector_type(8))) float v8f;

static constexpr int kRes  = 256;
static constexpr int kRank = 32;
static constexpr int kPix  = kRes * kRes;     // 65536 pixels per plane

// ---------------------------------------------------------------------------
// Pass 1: CHW -> HWC transpose, LDS-tiled (coalesced on both sides).
//   t[(y*256+x)*32 + c] = g[c*65536 + (y*256+x)]
// ---------------------------------------------------------------------------
__global__ void __launch_bounds__(1024)
triplane_transpose_kernel(const float* __restrict__ g0,
                          const float* __restrict__ g1,
                          const float* __restrict__ g2,
                          float* __restrict__ t0,
                          float* __restrict__ t1,
                          float* __restrict__ t2) {
    __shared__ float tile[32][33];                 // +1 pad: no bank conflicts
    const float* g = (blockIdx.y == 0) ? g0 : (blockIdx.y == 1) ? g1 : g2;
    float*       t = (blockIdx.y == 0) ? t0 : (blockIdx.y == 1) ? t1 : t2;

    const int pix0 = blockIdx.x * 32;
    const int tx = threadIdx.x;                    // 0..31 (lane)
    const int ty = threadIdx.y;                    // 0..31

    // Load: channel ty, pixels pix0+tx -> coalesced 128B per wave.
    tile[tx][ty] = g[ty * kPix + pix0 + tx];
    __syncthreads();
    // Store: pixel pix0+ty, channels tx -> coalesced 128B per wave.
    t[(pix0 + ty) * kRank + tx] = tile[ty][tx];
}

// ---------------------------------------------------------------------------
// Pass 2: gather + bilinear + triple product + WMMA rank-reduction + softplus
// ---------------------------------------------------------------------------
struct Plane { int base; float wx, wy; };

template <bool HWC>
__global__ void __launch_bounds__(256)
triplane_density_kernel(const float* __restrict__ pts,
                        const float* __restrict__ p0,
                        const float* __restrict__ p1,
                        const float* __restrict__ p2,
                        const float* __restrict__ aabb,
                        float* __restrict__ out) {
    const int lane  = threadIdx.x & 31;
    const int m     = lane & 15;                   // point-in-tile (row M)
    const int upper = lane >> 4;                   // 0: ranks 4k..4k+1, 1: 4k+2..4k+3
    const int wave  = (blockIdx.x * blockDim.x + threadIdx.x) >> 5;
    const int point = wave * 16 + m;

    // ---- normalize point to [-1,1] per aabb --------------------------------
    const float a0x = aabb[0], a0y = aabb[1], a0z = aabb[2];
    const float a1x = aabb[3], a1y = aabb[4], a1z = aabb[5];
    const float px = (pts[point * 3 + 0] - a0x) * (2.0f / (a1x - a0x)) - 1.0f;
    const float py = (pts[point * 3 + 1] - a0y) * (2.0f / (a1y - a0y)) - 1.0f;
    const float pz = (pts[point * 3 + 2] - a0z) * (2.0f / (a1z - a0z)) - 1.0f;

    // ---- per-plane bilinear setup (align_corners=True, clamp like ref) -----
    auto mk = [](float u, float v) -> Plane {      // u -> W axis, v -> H axis
        const float x = (u + 1.0f) * 0.5f * (float)(kRes - 1);
        const float y = (v + 1.0f) * 0.5f * (float)(kRes - 1);
        const float x0f = fminf(fmaxf(floorf(x), 0.0f), (float)(kRes - 2));
        const float y0f = fminf(fmaxf(floorf(y), 0.0f), (float)(kRes - 2));
        Plane pl;
        pl.wx = x - x0f;
        pl.wy = y - y0f;
        const int x0 = (int)x0f, y0 = (int)y0f;
        pl.base = HWC ? (y0 * kRes + x0) * kRank   // float index of (y0,x0) texel
                      : (y0 * kRes + x0);          // pixel index (CHW fallback)
        return pl;
    };
    const Plane pl0 = mk(px, py);                  // combo (0,1)
    const Plane pl1 = mk(px, pz);                  // combo (0,2)
    const Plane pl2 = mk(py, pz);                  // combo (1,2)

    // ---- bilinear sample of 2 consecutive ranks ----------------------------
    auto sample2 = [](const float* __restrict__ g, const Plane& pl, int r) -> v2f {
        v2f v00, v01, v10, v11;
        if (HWC) {
            const float* b = g + pl.base + r;      // ranks contiguous: b64 loads
            v00 = *(const v2f*)(b);
            v01 = *(const v2f*)(b + kRank);
            v10 = *(const v2f*)(b + kRes * kRank);
            v11 = *(const v2f*)(b + kRes * kRank + kRank);
        } else {                                    // CHW fallback (strided)
            const float* bA = g + pl.base + r * kPix;
            const float* bB = bA + kPix;
            v00 = (v2f){bA[0],        bB[0]};
            v01 = (v2f){bA[1],        bB[1]};
            v10 = (v2f){bA[kRes],     bB[kRes]};
            v11 = (v2f){bA[kRes + 1], bB[kRes + 1]};
        }
        const v2f wx = {pl.wx, pl.wx};
        const v2f wy = {pl.wy, pl.wy};
        const v2f top = v00 + wx * (v01 - v00);
        const v2f bot = v10 + wx * (v11 - v10);
        return top + wy * (bot - top);
    };

    // ---- triple product, rank-reduce via WMMA (B = ones) -------------------
    // A chunk k holds P[m, 4k + 2*upper + {0,1}] -> documented f32 16x4 A layout.
    // D[m,n] = sum_k A[m,k] (all 16 columns identical; column layout irrelevant).
    v8f acc = {};
    const v2f ones = {1.0f, 1.0f};
#pragma unroll
    for (int k = 0; k < 8; ++k) {
        const int r = 4 * k + 2 * upper;
        const v2f s0 = sample2(p0, pl0, r);
        const v2f s1 = sample2(p1, pl1, r);
        const v2f s2 = sample2(p2, pl2, r);
        const v2f prod = s0 * s1 * s2;
        acc = __builtin_amdgcn_wmma_f32_16x16x4_f32(
            /*neg_a=*/false, prod, /*neg_b=*/false, ones,
            /*c_mod=*/(short)0, acc, /*reuse_a=*/false, /*reuse_b=*/false);
    }

    // ---- mean + softplus + store -------------------------------------------
    // D row layout: VGPR j holds M=j in lanes 0-15 and M=j+8 in lanes 16-31.
    // Lane 0 stores points 0..7 of the tile, lane 16 stores points 8..15.
    if (m == 0) {
        const int tile_base = wave * 16 + upper * 8;
#pragma unroll
        for (int j = 0; j < 8; ++j) {
            const float x  = acc[j] * (1.0f / (float)kRank);
            const float sp = fmaxf(x, 0.0f) + log1pf(expf(-fabsf(x)));
            out[tile_base + j] = sp;
        }
    }
}

// ---------------------------------------------------------------------------
extern "C" void kernel_launch(void* const* d_in, const int* in_sizes, int n_in,
                              void* d_out, int out_size, void* d_ws, size_t ws_size,
                              hipStream_t stream) {
    const float* pts  = (const float*)d_in[0];   // [8192,256,3]
    const float* g0   = (const float*)d_in[1];   // [32,256,256]
    const float* g1   = (const float*)d_in[2];
    const float* g2   = (const float*)d_in[3];
    const float* aabb = (const float*)d_in[4];   // [2,3]
    float* out        = (float*)d_out;           // [8192,256,1]

    const int n_points = in_sizes[0] / 3;        // 2,097,152
    const int waves    = n_points / 16;
    const int threads  = 256;                    // 8 waves / block
    const int blocks   = (waves * 32 + threads - 1) / threads;

    const size_t plane_elems = (size_t)kRank * kPix;             // 2,097,152
    const size_t need        = 3 * plane_elems * sizeof(float);  // 24 MB

    if (ws_size >= need) {
        float* t0 = (float*)d_ws;
        float* t1 = t0 + plane_elems;
        float* t2 = t1 + plane_elems;
        triplane_transpose_kernel<<<dim3(kPix / 32, 3), dim3(32, 32), 0, stream>>>(
            g0, g1, g2, t0, t1, t2);
        triplane_density_kernel<true><<<blocks, threads, 0, stream>>>(
            pts, t0, t1, t2, aabb, out);
    } else {
        triplane_density_kernel<false><<<blocks, threads, 0, stream>>>(
            pts, g0, g1, g2, aabb, out);
    }
}